// RetinaNetFocalLoss_27865747816757
// MI455X (gfx1250) — compile-verified
//
#include <hip/hip_runtime.h>

// ---------------------------------------------------------------------------
// RetinaNet focal loss for MI455X (gfx1250, wave32).
// Pipeline (all on `stream`):
//   K0 init:     zero per-image accumulators in workspace
//   K1 match:    IoU argmax vs 64 targets (LDS-staged), matches[] to ws,
//                fused Huber numerator + match count (WMMA wave reduction)
//   K2 focal:    252MB streaming pass over clas_preds, B128 NT loads,
//                fused sigmoid/softplus (1 exp + 1 log + 1 rcp per element),
//                WMMA wave reduction -> per-block atomicAdd
//   K3 final:    per-image normalization max(4m,1)/max(m,1), mean over B
// ---------------------------------------------------------------------------

typedef __attribute__((ext_vector_type(2))) float v2f;
typedef __attribute__((ext_vector_type(4))) float v4f;
typedef __attribute__((ext_vector_type(8))) float v8f;

#define THREADS 256
#define ITER 4      // float4 chunks per thread in focal kernel
#define MAXT 128    // max targets stageable in LDS (T=64 here)

// Exact f32 wave(32) sum via V_WMMA_F32_16X16X4_F32:
// A[m][k]: lanes 0-15 hold {K0,K1}, lanes 16-31 hold {K2,K3}. With a={p,0}:
// A[m][0]=p_m, A[m][2]=p_{m+16}. B = all ones (4x16). D[m][n] = p_m + p_{m+16}.
// Lane L<16 holds D[0..7][L] in its 8 regs; lane L>=16 holds D[8..15][L-16].
// Summing own 8 regs then xor-16 shuffle-add yields the full wave sum in every lane.
__device__ __forceinline__ float wave_sum(float p) {
#if __has_builtin(__builtin_amdgcn_wmma_f32_16x16x4_f32)
    v2f a; a.x = p;    a.y = 0.0f;
    v2f b; b.x = 1.0f; b.y = 1.0f;
    v8f c = {0.f, 0.f, 0.f, 0.f, 0.f, 0.f, 0.f, 0.f};
    v8f d = __builtin_amdgcn_wmma_f32_16x16x4_f32(
        /*neg_a=*/false, a, /*neg_b=*/false, b,
        /*c_mod=*/(short)0, c, /*reuse_a=*/false, /*reuse_b=*/false);
    float s = ((d[0] + d[1]) + (d[2] + d[3])) + ((d[4] + d[5]) + (d[6] + d[7]));
    s += __shfl_xor(s, 16, 32);
    return s;
#else
    for (int off = 16; off > 0; off >>= 1) p += __shfl_xor(p, off, 32);
    return p;
#endif
}

// Block-wide sum -> one atomicAdd. Must be called by ALL threads (WMMA needs
// EXEC all-ones); inactive threads contribute 0.
__device__ __forceinline__ void block_atomic_add(float v, float* dst, float* lds8) {
    float w = wave_sum(v);
    const int lane = threadIdx.x & 31;
    const int wid  = threadIdx.x >> 5;
    if (lane == 0) lds8[wid] = w;
    __syncthreads();
    if (threadIdx.x == 0) {
        float t = 0.f;
        const int nw = blockDim.x >> 5;
        for (int i = 0; i < nw; ++i) t += lds8[i];
        atomicAdd(dst, t);
    }
    __syncthreads();
}

__global__ void init_kernel(float* acc, int n) {
    int i = blockIdx.x * blockDim.x + threadIdx.x;
    if (i < n) acc[i] = 0.0f;
}

// grid: (ceil(A/256), B), block: 256
__global__ void match_kernel(const float* __restrict__ bbox_preds, // B*A*4
                             const float* __restrict__ bbox_tgts,  // B*T*4 (tlbr)
                             const float* __restrict__ anchors,    // A*4  (cthw)
                             int*   __restrict__ matches,          // B*A
                             float* __restrict__ counts,           // B
                             float* __restrict__ bb_raw,           // B
                             int A, int T) {
    const int b = blockIdx.y;
    const int a = blockIdx.x * blockDim.x + threadIdx.x;

    __shared__ float t_tlx[MAXT], t_tly[MAXT], t_brx[MAXT], t_bry[MAXT];
    __shared__ float t_cx[MAXT], t_cy[MAXT], t_w[MAXT], t_h[MAXT], t_area[MAXT];
    __shared__ float lds8[8];

    for (int t = threadIdx.x; t < T; t += blockDim.x) {
        const float* bt = bbox_tgts + ((long long)b * T + t) * 4;
        float tlx = bt[0], tly = bt[1], brx = bt[2], bry = bt[3];
        t_tlx[t] = tlx; t_tly[t] = tly; t_brx[t] = brx; t_bry[t] = bry;
        float w = brx - tlx, h = bry - tly;
        t_cx[t] = (tlx + brx) * 0.5f;
        t_cy[t] = (tly + bry) * 0.5f;
        t_w[t] = w; t_h[t] = h;
        t_area[t] = w * h;
    }
    __syncthreads();

    float cnt = 0.f, hub = 0.f;
    if (a < A) {
        const v4f anc = *(const v4f*)(anchors + (size_t)a * 4);
        const float acx = anc[0], acy = anc[1], aw = anc[2], ah = anc[3];
        const float atlx = acx - aw * 0.5f, atly = acy - ah * 0.5f;
        const float abrx = acx + aw * 0.5f, abry = acy + ah * 0.5f;
        const float aarea = aw * ah;

        float best = -1.0f; int bidx = 0;
        for (int t = 0; t < T; ++t) {
            float iw = fminf(abrx, t_brx[t]) - fmaxf(atlx, t_tlx[t]);
            float ih = fminf(abry, t_bry[t]) - fmaxf(atly, t_tly[t]);
            iw = fmaxf(iw, 0.f); ih = fmaxf(ih, 0.f);
            float inter = iw * ih;
            float iou = inter / (aarea + t_area[t] - inter + 1e-8f);
            if (iou > best) { best = iou; bidx = t; }  // first-max (jnp.argmax)
        }
        int match = (best > 0.5f) ? bidx : ((best < 0.4f) ? -1 : -2);
        matches[(long long)b * A + a] = match;

        if (match >= 0) {
            cnt = 1.0f;
            const v4f bp = *(const v4f*)(bbox_preds + ((long long)b * A + a) * 4);
            const int g = match;
            // activ = concat((g.xy-anc.xy)/anc.wh, log(g.wh/anc.wh + 1e-8)) / SCALE
            float tcx  = (t_cx[g] - acx) / aw * 10.0f;
            float tcy  = (t_cy[g] - acy) / ah * 10.0f;
            float tszx = __logf(t_w[g] / aw + 1e-8f) * 5.0f;
            float tszy = __logf(t_h[g] / ah + 1e-8f) * 5.0f;
            float d0 = bp[0] - tcx, d1 = bp[1] - tcy, d2 = bp[2] - tszx, d3 = bp[3] - tszy;
            float a0 = fabsf(d0), a1 = fabsf(d1), a2 = fabsf(d2), a3 = fabsf(d3);
            hub  = (a0 < 1.f) ? 0.5f * d0 * d0 : a0 - 0.5f;
            hub += (a1 < 1.f) ? 0.5f * d1 * d1 : a1 - 0.5f;
            hub += (a2 < 1.f) ? 0.5f * d2 * d2 : a2 - 0.5f;
            hub += (a3 < 1.f) ? 0.5f * d3 * d3 : a3 - 0.5f;
        }
    }
    block_atomic_add(cnt, counts + b, lds8);
    block_atomic_add(hub, bb_raw + b, lds8);
}

// grid: (ceil(N4/(256*ITER)), B), block: 256. Requires (A*C) % 4 == 0 (C=80).
__global__ void focal_kernel(const float* __restrict__ clas_preds, // B*A*C
                             const int*   __restrict__ clas_tgts,  // B*T
                             const int*   __restrict__ matches,    // B*A
                             float* __restrict__ focal_raw,        // B
                             int A, int C, int T) {
    const int b = blockIdx.y;
    const long long N  = (long long)A * C;
    const long long N4 = N >> 2;
    const v4f* cp4     = (const v4f*)(clas_preds + (long long)b * N);
    const int* mrow    = matches + (long long)b * A;
    const int* ctrow   = clas_tgts + (long long)b * T;

    __shared__ float lds8[8];
    float acc = 0.0f;

    for (int it = 0; it < ITER; ++it) {
        long long ch = ((long long)blockIdx.x * ITER + it) * blockDim.x + threadIdx.x;
        if (ch >= N4) continue;
        v4f v = __builtin_nontemporal_load(cp4 + ch);   // streamed once: NT

        // C % 4 == 0 => an aligned float4 never straddles an anchor row.
        const long long idx0 = ch << 2;
        const int a  = (int)(idx0 / C);
        const int c0 = (int)(idx0 - (long long)a * C);

        const int m = mrow[a];
        if (m < -1) continue;                 // ignore zone (clas_mask == 0)
        const int cls    = (m >= 0) ? ctrow[m] : 0;  // 1..C when matched
        const int tgtIdx = cls - 1;           // -1 => all-zero one-hot

        #pragma unroll
        for (int j = 0; j < 4; ++j) {
            const float x = v[j];
            // shared exp: e = exp(-|x|) in (0,1]
            const float e  = __expf(-fabsf(x));
            const float sp = __logf(1.0f + e);          // log1p(e)
            const float sp_pos = fmaxf(x, 0.f) + sp;    // softplus(x)
            const float inv = 1.0f / (1.0f + e);
            const float ps  = (x >= 0.f) ? inv : 1.0f - inv;  // sigmoid(x)
            float w, bce;
            if (c0 + j == tgtIdx) {
                const float q = 1.0f - ps;
                w   = 0.75f * q * q;          // (1-ps)^2 * (1-ALPHA)
                bce = sp_pos - x;             // softplus(-x)
            } else {
                w   = 0.25f * ps * ps;        // ps^2 * ALPHA
                bce = sp_pos;                 // softplus(x)
            }
            acc += w * bce;
        }
    }
    block_atomic_add(acc, focal_raw + b, lds8);
}

__global__ void finalize_kernel(const float* __restrict__ counts,
                                const float* __restrict__ bb_raw,
                                const float* __restrict__ focal_raw,
                                float* __restrict__ out, int B) {
    if (blockIdx.x == 0 && threadIdx.x == 0) {
        float tot = 0.f;
        for (int b = 0; b < B; ++b) {
            const float mf = counts[b];
            tot += bb_raw[b]    / fmaxf(mf * 4.0f, 1.0f);
            tot += focal_raw[b] / fmaxf(mf,        1.0f);
        }
        out[0] = tot / (float)B;
    }
}

extern "C" void kernel_launch(void* const* d_in, const int* in_sizes, int n_in,
                              void* d_out, int out_size, void* d_ws, size_t ws_size,
                              hipStream_t stream) {
    const float* clas_preds = (const float*)d_in[0];
    const float* bbox_preds = (const float*)d_in[1];
    const float* bbox_tgts  = (const float*)d_in[2];
    const int*   clas_tgts  = (const int*)d_in[3];
    const float* anchors    = (const float*)d_in[4];

    const int A = in_sizes[4] / 4;              // 49152
    const int B = in_sizes[1] / (A * 4);        // 16
    const int T = in_sizes[3] / B;              // 64
    const int C = in_sizes[0] / (B * A);        // 80

    // Workspace layout: counts[B] | bb_raw[B] | focal_raw[B] | pad | matches[B*A]
    float* counts    = (float*)d_ws;
    float* bb_raw    = counts + B;
    float* focal_raw = bb_raw + B;
    const size_t moff = ((size_t)(3 * B) * sizeof(float) + 255) & ~(size_t)255;
    int* matches = (int*)((char*)d_ws + moff);

    // K0: zero accumulators
    init_kernel<<<1, 128, 0, stream>>>(counts, 3 * B);

    // K1: matching + huber numerator + match count
    dim3 g1((A + THREADS - 1) / THREADS, B);
    match_kernel<<<g1, THREADS, 0, stream>>>(bbox_preds, bbox_tgts, anchors,
                                             matches, counts, bb_raw, A, T);

    // K2: focal numerator (dominant 252MB streaming pass)
    const long long N4 = ((long long)A * C) >> 2;
    const long long bx = (N4 + (long long)THREADS * ITER - 1) / ((long long)THREADS * ITER);
    dim3 g2((unsigned)bx, B);
    focal_kernel<<<g2, THREADS, 0, stream>>>(clas_preds, clas_tgts, matches,
                                             focal_raw, A, C, T);

    // K3: normalize + mean
    finalize_kernel<<<1, 32, 0, stream>>>(counts, bb_raw, focal_raw, (float*)d_out, B);
}